// GATConv_10934986736300
// MI455X (gfx1250) — compile-verified
//
#include <hip/hip_runtime.h>
#include <math.h>

typedef __attribute__((ext_vector_type(2))) float v2f;
typedef __attribute__((ext_vector_type(8))) float v8f;

#define NNODES 50000
#define NEDGES 800000
#define INCH   256
#define OUTCH  64
#define LRELU_ALPHA 0.2f
#define NEG_BIG -9.0e15f

// workspace layout in float words
#define WH_OFF   0ull          // 50000*64
#define ES_OFF   3200000ull    // 50000
#define ED_OFF   3250000ull    // 50000
#define ELOG_OFF 3300000ull    // 800000 (logits, then exp weights)
#define PART_OFF 4100000ull    // 3125 block partials
#define GMAX_OFF 4103200ull
#define GSUM_OFF 4103201ull

#define EDGE_BLOCKS 3125       // 800000 / 256 exactly

// ---------------- init: zero accumulation output ----------------
__global__ void k_init(float* __restrict__ out, int n) {
  int i = blockIdx.x * blockDim.x + threadIdx.x;
  if (i < n) out[i] = 0.0f;
}

// ---------------- GEMM: wh = x @ W via V_WMMA_F32_16X16X4_F32 ----------------
// grid = (3125, 4), block = 32 (one wave per 16x16 tile of wh)
__global__ __launch_bounds__(32) void k_gemm(const float* __restrict__ x,
                                             const float* __restrict__ W,
                                             float* __restrict__ wh) {
  const int lane = threadIdx.x;          // wave32
  const int mloc = lane & 15;
  const int half = lane >> 4;
  const int kb   = half * 2;             // lanes 0-15: K 0,1 ; lanes 16-31: K 2,3
  const int row  = blockIdx.x * 16 + mloc;
  const int ncol = blockIdx.y * 16 + mloc;
  const float* __restrict__ arow = x + (size_t)row * INCH;

  v8f acc = {};
#pragma unroll 4
  for (int k = 0; k < INCH; k += 4) {
    v2f a, b;
    a.x = arow[k + kb];
    a.y = arow[k + kb + 1];
    b.x = W[(size_t)(k + kb) * OUTCH + ncol];
    b.y = W[(size_t)(k + kb + 1) * OUTCH + ncol];
    acc = __builtin_amdgcn_wmma_f32_16x16x4_f32(false, a, false, b,
                                                (short)0, acc, false, false);
  }
  // D layout: VGPR r -> lanes 0-15: M=r, lanes 16-31: M=8+r; N = lane&15
  const int rbase = blockIdx.x * 16 + half * 8;
#pragma unroll
  for (int r = 0; r < 8; ++r)
    wh[(size_t)(rbase + r) * OUTCH + ncol] = acc[r];
}

// ---------------- per-node attention scores ----------------
__global__ void k_escore(const float* __restrict__ wh, const float* __restrict__ a,
                         float* __restrict__ es, float* __restrict__ ed) {
  int i = blockIdx.x * blockDim.x + threadIdx.x;
  if (i >= NNODES) return;
  const float* __restrict__ r = wh + (size_t)i * OUTCH;
  float s1 = 0.0f, s2 = 0.0f;
#pragma unroll
  for (int f = 0; f < OUTCH; f += 4) {
    float4 v  = *(const float4*)(r + f);
    float4 a1 = *(const float4*)(a + f);
    float4 a2 = *(const float4*)(a + OUTCH + f);
    s1 += v.x * a1.x + v.y * a1.y + v.z * a1.z + v.w * a1.w;
    s2 += v.x * a2.x + v.y * a2.y + v.z * a2.z + v.w * a2.w;
  }
  es[i] = s1;
  ed[i] = s2;
}

// ---------------- edge logits + per-block max ----------------
__global__ void k_logits(const float* __restrict__ es, const float* __restrict__ ed,
                         const int* __restrict__ src, const int* __restrict__ dst,
                         float* __restrict__ elog, float* __restrict__ part) {
  __shared__ float red[256];
  int e = blockIdx.x * 256 + threadIdx.x;           // grid covers exactly NEDGES
  float v = es[src[e]] + ed[dst[e]];
  // leaky_relu then mask: result>0 iff v>0, and then equals v
  float lg = (v > 0.0f) ? v : NEG_BIG;
  elog[e] = lg;
  red[threadIdx.x] = lg;
  __syncthreads();
  for (int s = 128; s > 0; s >>= 1) {
    if (threadIdx.x < s) red[threadIdx.x] = fmaxf(red[threadIdx.x], red[threadIdx.x + s]);
    __syncthreads();
  }
  if (threadIdx.x == 0) part[blockIdx.x] = red[0];
}

__global__ void k_maxfinish(const float* __restrict__ part, float* __restrict__ gmax) {
  __shared__ float red[256];
  float m = NEG_BIG;
  for (int i = threadIdx.x; i < EDGE_BLOCKS; i += 256) m = fmaxf(m, part[i]);
  red[threadIdx.x] = m;
  __syncthreads();
  for (int s = 128; s > 0; s >>= 1) {
    if (threadIdx.x < s) red[threadIdx.x] = fmaxf(red[threadIdx.x], red[threadIdx.x + s]);
    __syncthreads();
  }
  if (threadIdx.x == 0) gmax[0] = red[0];
}

// ---------------- exp(e - max) + per-block sum ----------------
__global__ void k_expsum(float* __restrict__ elog, const float* __restrict__ gmax,
                         float* __restrict__ part) {
  __shared__ float red[256];
  int e = blockIdx.x * 256 + threadIdx.x;
  float w = expf(elog[e] - gmax[0]);                // NEG_BIG -> 0
  elog[e] = w;                                      // reuse buffer for exp weights
  red[threadIdx.x] = w;
  __syncthreads();
  for (int s = 128; s > 0; s >>= 1) {
    if (threadIdx.x < s) red[threadIdx.x] += red[threadIdx.x + s];
    __syncthreads();
  }
  if (threadIdx.x == 0) part[blockIdx.x] = red[0];
}

__global__ void k_sumfinish(const float* __restrict__ part, float* __restrict__ gsum) {
  __shared__ float red[256];
  float s0 = 0.0f;
  for (int i = threadIdx.x; i < EDGE_BLOCKS; i += 256) s0 += part[i];
  red[threadIdx.x] = s0;
  __syncthreads();
  for (int s = 128; s > 0; s >>= 1) {
    if (threadIdx.x < s) red[threadIdx.x] += red[threadIdx.x + s];
    __syncthreads();
  }
  if (threadIdx.x == 0) gsum[0] = red[0];
}

// ---------------- message scatter: out[dst] += att * wh[src] ----------------
// one thread per (edge, feature); 64 consecutive threads share one edge
__global__ void k_scatter(const float* __restrict__ wh, const float* __restrict__ expw,
                          const int* __restrict__ src, const int* __restrict__ dst,
                          const float* __restrict__ gsum, float* __restrict__ out) {
  long long tid = (long long)blockIdx.x * 256 + threadIdx.x;
  if (tid >= (long long)NEDGES * OUTCH) return;
  int e = (int)(tid >> 6);
  int f = (int)(tid & 63);
  float att = expw[e] * (1.0f / gsum[0]);
  float m = att * wh[(size_t)src[e] * OUTCH + f];
  atomicAdd(&out[(size_t)dst[e] * OUTCH + f], m);
}

// ---------------- ELU epilogue (in place on d_out) ----------------
__global__ void k_elu(float* __restrict__ out, int n) {
  int i = blockIdx.x * blockDim.x + threadIdx.x;
  if (i >= n) return;
  float v = out[i];
  out[i] = (v > 0.0f) ? v : (expf(v) - 1.0f);
}

extern "C" void kernel_launch(void* const* d_in, const int* in_sizes, int n_in,
                              void* d_out, int out_size, void* d_ws, size_t ws_size,
                              hipStream_t stream) {
  const float* x  = (const float*)d_in[0];
  const int*   ei = (const int*)d_in[1];      // [2, NEDGES]
  const float* W  = (const float*)d_in[2];
  const float* a  = (const float*)d_in[3];    // [1, 128]
  const int* src = ei;
  const int* dst = ei + NEDGES;

  float* ws   = (float*)d_ws;
  float* wh   = ws + WH_OFF;
  float* es   = ws + ES_OFF;
  float* ed   = ws + ED_OFF;
  float* elog = ws + ELOG_OFF;
  float* part = ws + PART_OFF;
  float* gmax = ws + GMAX_OFF;
  float* gsum = ws + GSUM_OFF;
  float* out  = (float*)d_out;

  const int nOut = NNODES * OUTCH;            // 3,200,000

  k_init<<<(nOut + 255) / 256, 256, 0, stream>>>(out, nOut);

  dim3 ggrid(NNODES / 16, OUTCH / 16);        // (3125, 4)
  k_gemm<<<ggrid, 32, 0, stream>>>(x, W, wh);

  k_escore<<<(NNODES + 255) / 256, 256, 0, stream>>>(wh, a, es, ed);

  k_logits<<<EDGE_BLOCKS, 256, 0, stream>>>(es, ed, src, dst, elog, part);
  k_maxfinish<<<1, 256, 0, stream>>>(part, gmax);

  k_expsum<<<EDGE_BLOCKS, 256, 0, stream>>>(elog, gmax, part);
  k_sumfinish<<<1, 256, 0, stream>>>(part, gsum);

  long long nScatter = (long long)NEDGES * OUTCH;          // 51.2M
  int sblocks = (int)((nScatter + 255) / 256);             // 200,000
  k_scatter<<<sblocks, 256, 0, stream>>>(wh, elog, src, dst, gsum, out);

  k_elu<<<(nOut + 255) / 256, 256, 0, stream>>>(out, nOut);
}